// SplatLatticeModule_25400436588637
// MI455X (gfx1250) — compile-verified
//
#include <hip/hip_runtime.h>
#include <stdint.h>

typedef __attribute__((ext_vector_type(2))) float v2f;
typedef __attribute__((ext_vector_type(8))) float v8f;

#define LATTICE_CAPACITY (1u << 20)
#define ROWLEN 65

// ---------------------------------------------------------------------------
// Kernel 1: zero the output table (harness poisons d_out with 0xAA; we must
// re-zero on every launch so graph replays stay correct).
// ---------------------------------------------------------------------------
__global__ __launch_bounds__(256) void zero_table_kernel(float* __restrict__ out, int n) {
  int i4 = (blockIdx.x * 256 + threadIdx.x) * 4;
  if (i4 + 3 < n) {
    float4 z; z.x = 0.f; z.y = 0.f; z.z = 0.f; z.w = 0.f;
    *(float4*)(out + i4) = z;
  } else {
    for (int k = 0; k < 4; ++k)
      if (i4 + k < n) out[i4 + k] = 0.f;
  }
}

// ---------------------------------------------------------------------------
// Kernel 2: permutohedral splat. One wave32 per point per iteration.
// All lanes compute the (cheap) lattice geometry redundantly; the 4x65
// outer product w (x) [values,1] is formed with V_WMMA_F32_16X16X4_F32
// (K=1 rank-1 trick, numerically identical to v_mul_f32) and scattered
// with coalesced 16-lane global_atomic_add_f32.
// ---------------------------------------------------------------------------
__global__ __launch_bounds__(256) void splat_kernel(
    const float* __restrict__ pos,
    const float* __restrict__ vals,
    float* __restrict__ table,
    int npoints)
{
  const int lane = (int)(threadIdx.x & 31u);
  const int wavesPerBlock = (int)(blockDim.x >> 5);
  const int wave   = (int)blockIdx.x * wavesPerBlock + ((int)threadIdx.x >> 5);
  const int nwaves = (int)gridDim.x * wavesPerBlock;

  // scale = inv_std / sqrt((i+1)(i+2)), inv_std = 4*sqrt(2/3)
  const float S0 = 2.30940107675850306f;   // 4*sqrt(2/3)/sqrt(2)
  const float S1 = 1.33333333333333333f;   // 4*sqrt(2/3)/sqrt(6)
  const float S2 = 0.94280904158206336f;   // 4*sqrt(2/3)/sqrt(12)

  for (int n = wave; n < npoints; n += nwaves) {
    float cf0 = pos[3 * n + 0] * S0;
    float cf1 = pos[3 * n + 1] * S1;
    float cf2 = pos[3 * n + 2] * S2;

    float elev[4];
    elev[0] = cf0 + cf1 + cf2;
    elev[1] = cf1 + cf2 - cf0;
    elev[2] = cf2 - 2.0f * cf1;
    elev[3] = -3.0f * cf2;

    float rem0[4], diff[4];
    float rdsumf = 0.f;
    #pragma unroll
    for (int i = 0; i < 4; ++i) {
      float rd = rintf(elev[i] * 0.25f);   // round-half-even, matches jnp.round
      rem0[i] = rd * 4.0f;
      rdsumf += rd;
      diff[i] = elev[i] - rem0[i];
    }
    const int rdsum = (int)rdsumf;

    int rank[4] = {0, 0, 0, 0};
    #pragma unroll
    for (int i = 0; i < 4; ++i) {
      #pragma unroll
      for (int j = 0; j < 4; ++j) {
        if (diff[j] > diff[i] || (diff[j] == diff[i] && j < i)) rank[i]++;
      }
      rank[i] += rdsum;
      if (rank[i] < 0)      { rank[i] += 4; rem0[i] += 4.f; }
      else if (rank[i] > 3) { rank[i] -= 4; rem0[i] -= 4.f; }
    }

    float delta[4];
    #pragma unroll
    for (int i = 0; i < 4; ++i) delta[i] = (elev[i] - rem0[i]) * 0.25f;

    // barycentric weights
    float b0 = 0.f, b1 = 0.f, b2 = 0.f, b3 = 0.f, b4 = 0.f;
    #pragma unroll
    for (int i = 0; i < 4; ++i) {
      int a = 3 - rank[i];
      float di = delta[i];
      b0 += (a == 0) ? di : 0.f;
      b1 += (a == 1) ? di : 0.f;
      b2 += (a == 2) ? di : 0.f;
      b3 += (a == 3) ? di : 0.f;
      b1 -= (a == 0) ? di : 0.f;
      b2 -= (a == 1) ? di : 0.f;
      b3 -= (a == 2) ? di : 0.f;
      b4 -= (a == 3) ? di : 0.f;
    }
    b0 += 1.0f + b4;
    const float w0 = b0, w1 = b1, w2 = b2, w3 = b3;

    // hashes for the 4 simplex vertices (uint32 wraparound, % 2^20)
    const int ir0 = (int)rem0[0], ir1 = (int)rem0[1], ir2 = (int)rem0[2];
    uint32_t h[4];
    #pragma unroll
    for (int r = 0; r < 4; ++r) {
      int o0 = (rank[0] < 4 - r) ? r : r - 4;
      int o1 = (rank[1] < 4 - r) ? r : r - 4;
      int o2 = (rank[2] < 4 - r) ? r : r - 4;
      uint32_t hh = (uint32_t)(ir0 + o0) * 2654435761u
                  + (uint32_t)(ir1 + o1) * 805459861u
                  + (uint32_t)(ir2 + o2) * 3674653429u;
      h[r] = hh & (LATTICE_CAPACITY - 1u);
    }

    // --- WMMA outer product: D[m,j] = w[m] * vh_chunk[j], K=1 ---
    // A (16x4 f32): lanes 0-15 hold rows M=lane, VGPR0 = K=0, VGPR1 = K=1;
    // lanes 16-31 hold K=2/3. Only A[m<4, 0] = w[m] is nonzero.
    const float wsel = (lane == 0) ? w0 : (lane == 1) ? w1 :
                       (lane == 2) ? w2 : (lane == 3) ? w3 : 0.0f;
    v2f A; A[0] = wsel; A[1] = 0.f;
    v8f Cz = {};

    const float* vrow = vals + (size_t)n * 64;
    const size_t base0 = (size_t)h[0] * ROWLEN;
    const size_t base1 = (size_t)h[1] * ROWLEN;
    const size_t base2 = (size_t)h[2] * ROWLEN;
    const size_t base3 = (size_t)h[3] * ROWLEN;

    #pragma unroll
    for (int c = 0; c < 4; ++c) {
      // B (4x16 f32): row K=0 = values chunk in lanes 0-15 of VGPR0; rest zero.
      float bv = (lane < 16) ? vrow[c * 16 + lane] : 0.f;
      v2f B; B[0] = bv; B[1] = 0.f;
      v8f D = __builtin_amdgcn_wmma_f32_16x16x4_f32(
          /*neg_a=*/false, A, /*neg_b=*/false, B,
          /*c_mod=*/(short)0, Cz, /*reuse_a=*/false, /*reuse_b=*/false);
      // D VGPR r, lanes 0-15 = w[r] * vrow[c*16 + lane]; coalesced scatter.
      if (lane < 16) {
        int col = c * 16 + lane;
        atomicAdd(table + base0 + col, D[0]);
        atomicAdd(table + base1 + col, D[1]);
        atomicAdd(table + base2 + col, D[2]);
        atomicAdd(table + base3 + col, D[3]);
      }
    }
    // homogeneous coordinate: contrib = w[r] * 1.0
    if (lane < 4) {
      uint32_t hs = (lane == 0) ? h[0] : (lane == 1) ? h[1] :
                    (lane == 2) ? h[2] : h[3];
      atomicAdd(table + (size_t)hs * ROWLEN + 64, wsel);
    }
  }
}

// ---------------------------------------------------------------------------
extern "C" void kernel_launch(void* const* d_in, const int* in_sizes, int n_in,
                              void* d_out, int out_size, void* d_ws, size_t ws_size,
                              hipStream_t stream) {
  const float* pos  = (const float*)d_in[0];   // [N,3] f32
  const float* vals = (const float*)d_in[1];   // [N,64] f32
  float* table = (float*)d_out;                // [2^20, 65] f32
  const int npoints = in_sizes[0] / 3;

  // Zero the table (required every launch: d_out poisoned once, then replayed).
  int n4blocks = (out_size / 4 + 255) / 256 + 1;
  zero_table_kernel<<<n4blocks, 256, 0, stream>>>(table, out_size);

  // Splat: 2048 blocks x 8 waves = 16384 waves, ~16 points per wave.
  splat_kernel<<<2048, 256, 0, stream>>>(pos, vals, table, npoints);
}